// DialogueGCN_34282428957140
// MI455X (gfx1250) — compile-verified
//
#include <hip/hip_runtime.h>
#include <hip/hip_bf16.h>
#include <stdint.h>

#define NS    10          // speakers / nodes
#define D     1024        // hidden dim
#define NPAD  16          // padded M for WMMA tiles
#define NEDGE 100         // all-pairs edges
#define KCH   4           // K-chunks for the edge GEMV
#define KLEN  (D / KCH)   // 256

typedef __attribute__((ext_vector_type(2))) float v2f;
typedef __attribute__((ext_vector_type(4))) float v4f;
typedef __attribute__((ext_vector_type(8))) float v8f;

// ---------------------------------------------------------------------------
// Pad X [10,1024] -> Xpad [16,1024] (rows 10..15 = 0). ws is poisoned, so the
// pad rows MUST be written.
// ---------------------------------------------------------------------------
__global__ __launch_bounds__(256)
void prep_xpad(const float* __restrict__ X, float* __restrict__ Xpad) {
    int idx = blockIdx.x * 256 + threadIdx.x;      // < 16384
    int row = idx >> 10;
    Xpad[idx] = (row < NS) ? X[idx] : 0.0f;
}

// ---------------------------------------------------------------------------
// C[16,1024] = A[16,1024] @ B[1024,1024] (+ bias on real rows), f32 WMMA.
// One wave per 16x16 output tile; grid.x = 64 column tiles.
// A-frag layout (16x4 f32): lanes 0-15 hold K={k0,k0+1}, lanes 16-31 K={k0+2,k0+3}.
// C/D layout: VGPR r -> M=r (lanes 0-15) / M=r+8 (lanes 16-31), N=lane&15.
// ---------------------------------------------------------------------------
__global__ __launch_bounds__(32)
void gemm16_wmma_f32(const float* __restrict__ A, const float* __restrict__ B,
                     float* __restrict__ C, const float* __restrict__ bias,
                     int mReal) {
    const int lane = threadIdx.x;
    const int cb   = blockIdx.x * 16;
    const int mA   = lane & 15;
    const int hi   = lane >> 4;
    const int kA   = hi * 2;

    v8f acc = {0.f, 0.f, 0.f, 0.f, 0.f, 0.f, 0.f, 0.f};
    for (int k0 = 0; k0 < D; k0 += 4) {
        v2f a = { A[(size_t)mA * D + k0 + kA],
                  A[(size_t)mA * D + k0 + kA + 1] };
        v2f b = { B[(size_t)(k0 + hi) * D + cb + mA],
                  B[(size_t)(k0 + 2 + hi) * D + cb + mA] };
        acc = __builtin_amdgcn_wmma_f32_16x16x4_f32(
                  false, a, false, b, (short)0, acc, false, false);
    }
    const int n  = lane & 15;
    const int mb = hi * 8;
    for (int r = 0; r < 8; ++r) {
        int   m = mb + r;
        float v = acc[r];
        if (bias != nullptr && m < mReal) v += bias[cb + n];
        C[(size_t)m * D + cb + n] = v;   // pad rows store exact 0 (A rows are 0)
    }
}

// ---------------------------------------------------------------------------
// Attention scores -> softmax -> per-edge etype / mean-normalized weight w_e.
// Single block, 128 threads. Integer LDS atomics only (deterministic).
// ---------------------------------------------------------------------------
__global__ __launch_bounds__(128)
void attn_edges(const float* __restrict__ q, const float* __restrict__ kmat,
                const int* __restrict__ speaker,
                float* __restrict__ w_e, int* __restrict__ etype) {
    __shared__ float at[NS][NS];
    __shared__ int   etl[NEDGE];
    __shared__ int   cnt[2 * NS * NS * NS];        // R*N = 2000 segments
    const int tid = threadIdx.x;
    const int i = tid / NS, j = tid % NS;

    for (int s = tid; s < 2 * NS * NS * NS; s += blockDim.x) cnt[s] = 0;

    if (tid < NEDGE) {
        float sc = 0.f;
        for (int d = 0; d < D; ++d)
            sc = fmaf(q[(size_t)i * D + d], kmat[(size_t)j * D + d], sc);
        at[i][j] = sc * (1.0f / 32.0f);            // 1/sqrt(1024)
    }
    __syncthreads();

    if (tid < NS) {                                 // row softmax
        float m = at[tid][0];
        for (int c = 1; c < NS; ++c) m = fmaxf(m, at[tid][c]);
        float e[NS]; float s = 0.f;
        for (int c = 0; c < NS; ++c) { e[c] = __expf(at[tid][c] - m); s += e[c]; }
        float inv = 1.0f / s;
        for (int c = 0; c < NS; ++c) at[tid][c] = e[c] * inv;
    }
    __syncthreads();

    if (tid < NEDGE) {
        int et = 2 * (speaker[i] * NS + speaker[j]) + ((i >= j) ? 1 : 0);
        etl[tid] = et;
        atomicAdd(&cnt[et * NS + j], 1);
    }
    __syncthreads();

    if (tid < NEDGE) {
        int et = etl[tid];
        etype[tid] = et;
        w_e[tid]   = at[i][j] / (float)cnt[et * NS + j];
    }
}

// ---------------------------------------------------------------------------
// Dominant kernel: per-edge GEMV against rgcn_weight[etype] (read-once data).
// grid = NEDGE*KCH blocks; each block streams a 1 MiB K-chunk of one 4 MiB
// relation matrix with non-temporal 16B loads (don't pollute 192MB L2).
// part[blockIdx][0..1023] = w_e * (x_chunk @ W_chunk).
// ---------------------------------------------------------------------------
__global__ __launch_bounds__(256)
void edge_gemv(const float* __restrict__ X, const float* __restrict__ W,
               const float* __restrict__ w_e, const int* __restrict__ etype,
               float* __restrict__ part) {
    __shared__ float xs[KLEN];
    const int e   = blockIdx.x >> 2;               // edge 0..99
    const int kc  = blockIdx.x & 3;                // K-chunk 0..3
    const int src = e / NS;
    const int t   = etype[e];
    const float we = w_e[e];
    const int d0  = kc * KLEN;
    const int tid = threadIdx.x;

    xs[tid] = X[(size_t)src * D + d0 + tid];
    __syncthreads();

    const float* Wp = W + (size_t)t * D * D + (size_t)d0 * D + tid * 4;
    float ax = 0.f, ay = 0.f, az = 0.f, aw = 0.f;
    for (int dd = 0; dd < KLEN; ++dd) {
        v4f w4 = __builtin_nontemporal_load((const v4f*)(Wp + (size_t)dd * D));
        float xv = xs[dd];
        ax = fmaf(xv, w4.x, ax);
        ay = fmaf(xv, w4.y, ay);
        az = fmaf(xv, w4.z, az);
        aw = fmaf(xv, w4.w, aw);
    }
    float* o = part + (size_t)blockIdx.x * D + tid * 4;
    o[0] = we * ax; o[1] = we * ay; o[2] = we * az; o[3] = we * aw;
}

// ---------------------------------------------------------------------------
// out[j] = (X@rgcn_root + bias)[j] + sum over edges (i,j), k-chunks. Rows 10..15
// of outpad stay exactly 0 (written by the GEMM, untouched here).
// ---------------------------------------------------------------------------
__global__ __launch_bounds__(256)
void reduce_out(const float* __restrict__ part, float* __restrict__ outpad) {
    int idx = blockIdx.x * 256 + threadIdx.x;      // < 10240
    int j = idx >> 10, f = idx & (D - 1);
    float s = outpad[(size_t)j * D + f];
    for (int i = 0; i < NS; ++i) {
        size_t base = ((size_t)(i * NS + j) * KCH) * D + f;
        s += part[base] + part[base + D] + part[base + 2 * (size_t)D]
           + part[base + 3 * (size_t)D];
    }
    outpad[(size_t)j * D + f] = s;
}

// sum_out[f] = sum_i out[i][f]  (all-pairs graph collapses agg to this)
__global__ __launch_bounds__(256)
void sum_rows(const float* __restrict__ outpad, float* __restrict__ sum_out) {
    int f = blockIdx.x * 256 + threadIdx.x;        // < 1024
    float s = 0.f;
    for (int i = 0; i < NS; ++i) s += outpad[(size_t)i * D + f];
    sum_out[f] = s;
}

// t2 partials: sum_out @ gcn_rel_w split into 16 K-chunks (deterministic).
__global__ __launch_bounds__(256)
void gemv_rel(const float* __restrict__ sum_out, const float* __restrict__ G,
              float* __restrict__ t2p) {
    __shared__ float xs[64];
    const int d0  = blockIdx.x * 64;
    const int tid = threadIdx.x;
    if (tid < 64) xs[tid] = sum_out[d0 + tid];
    __syncthreads();

    const float* Gp = G + (size_t)d0 * D + tid * 4;
    float ax = 0.f, ay = 0.f, az = 0.f, aw = 0.f;
    for (int dd = 0; dd < 64; ++dd) {
        v4f g = __builtin_nontemporal_load((const v4f*)(Gp + (size_t)dd * D));
        float xv = xs[dd];
        ax = fmaf(xv, g.x, ax);
        ay = fmaf(xv, g.y, ay);
        az = fmaf(xv, g.z, az);
        aw = fmaf(xv, g.w, aw);
    }
    float* o = t2p + (size_t)blockIdx.x * D + tid * 4;
    o[0] = ax; o[1] = ay; o[2] = az; o[3] = aw;
}

// d_out[i] = [ t1[i] + (sum t2p + rel_b) , X[i] ]   -> [10, 2048]
__global__ __launch_bounds__(256)
void finalize(const float* __restrict__ t1, const float* __restrict__ t2p,
              const float* __restrict__ rel_b, const float* __restrict__ X,
              float* __restrict__ out) {
    int idx = blockIdx.x * 256 + threadIdx.x;      // < 10240
    int i = idx >> 10, f = idx & (D - 1);
    float t2 = rel_b[f];
    for (int kc = 0; kc < 16; ++kc) t2 += t2p[(size_t)kc * D + f];
    out[(size_t)i * 2 * D + f]     = t1[(size_t)i * D + f] + t2;
    out[(size_t)i * 2 * D + D + f] = X[(size_t)i * D + f];
}

// ---------------------------------------------------------------------------
extern "C" void kernel_launch(void* const* d_in, const int* in_sizes, int n_in,
                              void* d_out, int out_size, void* d_ws, size_t ws_size,
                              hipStream_t stream) {
    const float* X         = (const float*)d_in[0];
    const int*   speaker   = (const int*)  d_in[1];
    const float* Wq        = (const float*)d_in[2];
    const float* Wk        = (const float*)d_in[3];
    const float* rgcn_w    = (const float*)d_in[4];
    const float* rgcn_root = (const float*)d_in[5];
    const float* rgcn_bias = (const float*)d_in[6];
    const float* gcn_rel_w = (const float*)d_in[7];
    const float* gcn_rel_b = (const float*)d_in[8];
    const float* gcn_root_w= (const float*)d_in[9];
    float* out = (float*)d_out;

    float* w    = (float*)d_ws;
    float* Xpad = w;                    // 16*1024
    float* q    = w + 16384;            // 16*1024
    float* kmat = w + 32768;            // 16*1024
    float* outp = w + 49152;            // 16*1024
    float* t1   = w + 65536;            // 16*1024
    float* sumo = w + 81920;            // 1024
    float* t2p  = w + 82944;            // 16*1024
    float* part = w + 99328;            // 400*1024
    float* wec  = w + 508928;           // 100
    int*   etyp = (int*)(w + 509056);   // 100

    // 1. pad X for WMMA tiles
    prep_xpad<<<64, 256, 0, stream>>>(X, Xpad);
    // 2. q, k, out_init = X@rgcn_root + bias  (f32 WMMA, 64 col-tiles each)
    gemm16_wmma_f32<<<64, 32, 0, stream>>>(Xpad, Wq,        q,    nullptr,   NS);
    gemm16_wmma_f32<<<64, 32, 0, stream>>>(Xpad, Wk,        kmat, nullptr,   NS);
    gemm16_wmma_f32<<<64, 32, 0, stream>>>(Xpad, rgcn_root, outp, rgcn_bias, NS);
    // 3. softmax attention -> per-edge weights / relation ids
    attn_edges<<<1, 128, 0, stream>>>(q, kmat, speaker, wec, etyp);
    // 4. dominant 400 MiB relation-weight stream (NT loads)
    edge_gemv<<<NEDGE * KCH, 256, 0, stream>>>(X, rgcn_w, wec, etyp, part);
    // 5. deterministic segment reduction into out
    reduce_out<<<40, 256, 0, stream>>>(part, outp);
    // 6. agg collapses to a single row-sum
    sum_rows<<<4, 256, 0, stream>>>(outp, sumo);
    // 7. sum_out @ gcn_rel_w (partials), out @ gcn_root_w (WMMA)
    gemv_rel<<<16, 256, 0, stream>>>(sumo, gcn_rel_w, t2p);
    gemm16_wmma_f32<<<64, 32, 0, stream>>>(outp, gcn_root_w, t1, nullptr, NS);
    // 8. assemble [x2 | X]
    finalize<<<40, 256, 0, stream>>>(t1, t2p, gcn_rel_b, X, out);
}